// STGCNBlock_7198365188831
// MI455X (gfx1250) — compile-verified
//
#include <hip/hip_runtime.h>
#include <cstddef>

// ---------------- problem constants (from reference) ----------------
#define NJOINT 25          // V
#define CIN    64
#define COUT   64
#define VC     (NJOINT * COUT)      // 1600 flattened BN channels
#define NFRM   (64 * 300)           // B*T = 19200
#define BN_EPS 1e-5f

// gcn_gemm geometry
#define WPB        4                // waves per block (128 threads)
#define FPW        8                // frames per wave
#define BLOCKS_B   600              // 600*4 waves * 8 frames = 19200

// BN reduction geometry
#define NCHUNK     60
#define CHUNKROWS  320              // 60*320 = 19200

// workspace layout (float offsets)
#define WS_AHAT   0                               // 32x32 padded A_hat
#define WS_PSUM   1024                            // [NCHUNK][VC]
#define WS_PSQ    (WS_PSUM + NCHUNK * VC)         // [NCHUNK][VC]
#define WS_SCALE  (WS_PSQ  + NCHUNK * VC)         // [VC]  (16B aligned)
#define WS_SHIFT  (WS_SCALE + VC)                 // [VC]  (16B aligned)

typedef __attribute__((ext_vector_type(2))) float v2f;
typedef __attribute__((ext_vector_type(8))) float v8f;

// ---------------------------------------------------------------
// Kernel 1: build padded 32x32 normalized adjacency with self loops.
// A_hat[t][s] = sum_{edges (s->t)} dinv[s]*dinv[t]  (+ dinv[v]^2 on diag)
// ---------------------------------------------------------------
__global__ void stgcn_build_ahat(const int* __restrict__ edge_index, int E,
                                 float* __restrict__ ws) {
  if (threadIdx.x != 0 || blockIdx.x != 0) return;
  const int* row = edge_index;       // sources
  const int* col = edge_index + E;   // targets
  float deg[NJOINT];
  for (int v = 0; v < NJOINT; ++v) deg[v] = 1.0f;          // self loop
  for (int e = 0; e < E; ++e) deg[col[e]] += 1.0f;
  float dinv[NJOINT];
  for (int v = 0; v < NJOINT; ++v) dinv[v] = rsqrtf(deg[v]);
  float* A = ws + WS_AHAT;
  for (int i = 0; i < 32 * 32; ++i) A[i] = 0.0f;
  for (int e = 0; e < E; ++e) {
    int s = row[e], t = col[e];
    A[t * 32 + s] += dinv[s] * dinv[t];
  }
  for (int v = 0; v < NJOINT; ++v) A[v * 32 + v] += dinv[v] * dinv[v];
}

// ---------------------------------------------------------------
// Kernel 2: per-frame  H = relu( A_hat(32x32) * (X(32x64) * W(64x64)) + b )
// One wave32 per frame.  f32 WMMA 16x16x4 throughout.
//
// ISA f32 fragment layouts (CDNA5 ISA):
//   A 16x4 : lane l -> row m=l%16, VGPR j holds k = j + 2*(l/16)
//   B 4x16 : lane l -> col n=l%16, VGPR j holds k = j + 2*(l/16)
//   C/D    : lane l -> col n=l%16, VGPR r holds row m = r + 8*(l/16)
// All k-pairs are stored contiguously as float2 in LDS so each fragment
// is a single aligned ds_load_b64 into an even register pair.
// ---------------------------------------------------------------
#define WSTR 76   // wpair stride (float2)
#define PSTR 80   // pwpair stride (float2): 2*80*2 dwords == 320 = 0 mod 64? (160 dw/row -> halves +32 banks, disjoint)
#define ASTR 17   // apair stride (float2)

__launch_bounds__(WPB * 32, 1)
__global__ void stgcn_gcn_gemm(const float* __restrict__ x,
                               const float* __restrict__ W,
                               const float* __restrict__ bias,
                               const float* __restrict__ ws,
                               float* __restrict__ out) {
  __shared__ v2f wpair[CIN / 2][WSTR];        // {W[2p][d], W[2p+1][d]}
  __shared__ v2f apair[32][ASTR];             // {A[r][2p], A[r][2p+1]}
  __shared__ v2f pwbuf[WPB][16][PSTR];        // per-wave X*W staging, k-paired
  __shared__ float bbuf[COUT];

  const int tid = threadIdx.x;

  // stage W as k-pairs: wpair[p][d] = {W[2p][d], W[2p+1][d]}
  for (int i = tid; i < (CIN / 2) * COUT; i += WPB * 32) {
    const int p = i >> 6, d = i & 63;
    v2f val;
    val.x = W[(2 * p) * COUT + d];
    val.y = W[(2 * p + 1) * COUT + d];
    wpair[p][d] = val;
  }
  // stage A_hat as k-pairs within each row (rows are contiguous in ws)
  for (int i = tid; i < 32 * 16; i += WPB * 32) {
    apair[i >> 4][i & 15] = ((const v2f*)(ws + WS_AHAT))[i];
  }
  if (tid < COUT) bbuf[tid] = bias[tid];
  __syncthreads();

  const int wave = tid >> 5;
  const int lane = tid & 31;
  const int half = lane >> 4;      // 0: lanes 0-15, 1: lanes 16-31
  const int ln   = lane & 15;
  v2f (*pw)[PSTR] = pwbuf[wave];

  const int waveGlobal = blockIdx.x * WPB + wave;

  for (int f = 0; f < FPW; ++f) {
    const int frame = waveGlobal + f * (BLOCKS_B * WPB);     // 0..19199, exact cover
    const float* xf = x + (size_t)frame * (NJOINT * CIN);

    // ---------------- stage 1: PW = X * W  (K = 64 -> 16 wmma steps) ----------
#pragma unroll
    for (int mt = 0; mt < 2; ++mt) {
      v8f acc[4];
#pragma unroll
      for (int nt = 0; nt < 4; ++nt) acc[nt] = (v8f)(0.0f);

      const int m = mt * 16 + ln;                    // X row this lane supplies
      const bool mvalid = (m < NJOINT);
      const v2f* xrow = (const v2f*)(xf + m * CIN);  // 8B-aligned k-pairs
#pragma unroll
      for (int ks = 0; ks < 16; ++ks) {
        const int kp = 2 * ks + half;                // k-pair index
        v2f a = (v2f)(0.0f);
        if (mvalid) a = xrow[kp];
#pragma unroll
        for (int nt = 0; nt < 4; ++nt) {
          v2f b = wpair[kp][nt * 16 + ln];
          acc[nt] = __builtin_amdgcn_wmma_f32_16x16x4_f32(
              false, a, false, b, (short)0, acc[nt], false, false);
        }
      }
      // spill C/D-layout tile to LDS as k-pairs (rows r, r+1 live in this lane)
#pragma unroll
      for (int nt = 0; nt < 4; ++nt)
#pragma unroll
        for (int r = 0; r < 8; r += 2) {
          v2f val;
          val.x = acc[nt][r];
          val.y = acc[nt][r + 1];
          pw[mt * 8 + half * 4 + (r >> 1)][nt * 16 + ln] = val;
        }
    }

    // ---------------- stage 2: H = A_hat * PW  (K = 32 -> 8 wmma steps) -------
#pragma unroll
    for (int mt = 0; mt < 2; ++mt) {
      v8f acc[4];
#pragma unroll
      for (int nt = 0; nt < 4; ++nt) acc[nt] = (v8f)(0.0f);

#pragma unroll
      for (int ks = 0; ks < 8; ++ks) {
        const int kp = 2 * ks + half;
        v2f a = apair[mt * 16 + ln][kp];
#pragma unroll
        for (int nt = 0; nt < 4; ++nt) {
          v2f b = pw[kp][nt * 16 + ln];
          acc[nt] = __builtin_amdgcn_wmma_f32_16x16x4_f32(
              false, a, false, b, (short)0, acc[nt], false, false);
        }
      }
      // epilogue: + bias, relu, store in [frame][v*64 + c] layout (== final layout)
      float* of = out + (size_t)frame * VC;
#pragma unroll
      for (int nt = 0; nt < 4; ++nt) {
        const int colb = nt * 16 + ln;
        const float bb = bbuf[colb];
#pragma unroll
        for (int r = 0; r < 8; ++r) {
          const int v = mt * 16 + half * 8 + r;
          if (v < NJOINT) {
            float val = acc[nt][r] + bb;
            of[v * COUT + colb] = val > 0.0f ? val : 0.0f;
          }
        }
      }
    }
  }
}

// ---------------------------------------------------------------
// Kernel 3: deterministic partial BN stats. grid = (7, NCHUNK), block = 256.
// Reads d_out immediately after the GEMM: largely L2-resident (192 MB L2).
// ---------------------------------------------------------------
__global__ void stgcn_bn_partial(const float* __restrict__ out,
                                 float* __restrict__ ws) {
  const int ch = blockIdx.x * blockDim.x + threadIdx.x;
  if (ch >= VC) return;
  const int chunk = blockIdx.y;
  const float* p = out + (size_t)chunk * CHUNKROWS * VC + ch;
  float s = 0.0f, ss = 0.0f;
  for (int i = 0; i < CHUNKROWS; ++i) {
    float v = p[(size_t)i * VC];
    s += v;
    ss += v * v;
  }
  ws[WS_PSUM + chunk * VC + ch] = s;
  ws[WS_PSQ + chunk * VC + ch] = ss;
}

// ---------------------------------------------------------------
// Kernel 4: combine partials (fp64), produce per-channel scale/shift.
// ---------------------------------------------------------------
__global__ void stgcn_bn_finalize(const float* __restrict__ gamma,
                                  const float* __restrict__ beta,
                                  float* __restrict__ ws) {
  const int ch = blockIdx.x * blockDim.x + threadIdx.x;
  if (ch >= VC) return;
  double s = 0.0, ss = 0.0;
  for (int c = 0; c < NCHUNK; ++c) {
    s += (double)ws[WS_PSUM + c * VC + ch];
    ss += (double)ws[WS_PSQ + c * VC + ch];
  }
  const double n = (double)NFRM;
  const double mean = s / n;
  const double var = ss / n - mean * mean;           // biased, like torch BN
  const float scale = gamma[ch] * rsqrtf((float)var + BN_EPS);
  const float shift = beta[ch] - (float)mean * scale;
  ws[WS_SCALE + ch] = scale;
  ws[WS_SHIFT + ch] = shift;
}

// ---------------------------------------------------------------
// Kernel 5: in-place normalize, float4-vectorized (VC % 4 == 0).
// total elems = 30,720,000 -> 7,680,000 float4 -> 30000 blocks x 256.
// ---------------------------------------------------------------
__global__ void stgcn_bn_apply(float* __restrict__ out,
                               const float* __restrict__ ws) {
  const size_t idx = ((size_t)blockIdx.x * blockDim.x + threadIdx.x) * 4;
  float4 h = *(const float4*)(out + idx);
  const int ch = (int)(idx % VC);                    // multiple of 4
  const float4 sc = *(const float4*)(ws + WS_SCALE + ch);
  const float4 sh = *(const float4*)(ws + WS_SHIFT + ch);
  h.x = h.x * sc.x + sh.x;
  h.y = h.y * sc.y + sh.y;
  h.z = h.z * sc.z + sh.z;
  h.w = h.w * sc.w + sh.w;
  *(float4*)(out + idx) = h;
}

// ---------------------------------------------------------------
extern "C" void kernel_launch(void* const* d_in, const int* in_sizes, int n_in,
                              void* d_out, int out_size, void* d_ws, size_t ws_size,
                              hipStream_t stream) {
  const float* x     = (const float*)d_in[0];   // [64,300,25,64]
  const float* W     = (const float*)d_in[1];   // [64,64]
  const float* b     = (const float*)d_in[2];   // [64]
  const float* gamma = (const float*)d_in[3];   // [1600]
  const float* beta  = (const float*)d_in[4];   // [1600]
  const int*   ei    = (const int*)d_in[5];     // [2,E]
  float* out = (float*)d_out;                   // [64,300,1600]
  float* ws  = (float*)d_ws;
  const int E = in_sizes[5] / 2;

  hipLaunchKernelGGL(stgcn_build_ahat, dim3(1), dim3(64), 0, stream, ei, E, ws);
  hipLaunchKernelGGL(stgcn_gcn_gemm, dim3(BLOCKS_B), dim3(WPB * 32), 0, stream,
                     x, W, b, ws, out);
  hipLaunchKernelGGL(stgcn_bn_partial, dim3((VC + 255) / 256, NCHUNK), dim3(256),
                     0, stream, out, ws);
  hipLaunchKernelGGL(stgcn_bn_finalize, dim3((VC + 255) / 256), dim3(256),
                     0, stream, gamma, beta, ws);
  hipLaunchKernelGGL(stgcn_bn_apply, dim3((NFRM * VC) / 4 / 256), dim3(256),
                     0, stream, out, ws);
}